// MultiHeadAttn_10118942949614
// MI455X (gfx1250) — compile-verified
//
#include <hip/hip_runtime.h>

// ---------------------------------------------------------------------------
// MHA forward for MI455X (gfx1250, wave32), fp32 via V_WMMA_F32_16X16X4_F32.
// Outputs (concat flat): out [B,S,D] then masked scores [H,B,S,S].
// GEMM waves compute 32x64 tiles (2 M-subtiles) to halve B-fragment loads
// per WMMA; attention is flash-fused so p_attn never touches HBM.
// ---------------------------------------------------------------------------

typedef float v2f __attribute__((ext_vector_type(2)));
typedef float v8f __attribute__((ext_vector_type(8)));

constexpr int    kD    = 1024;   // d_model
constexpr int    kH    = 16;     // heads
constexpr int    kDk   = 64;     // head dim
constexpr int    kB    = 4;      // batch
constexpr int    kS    = 1024;   // seq
constexpr int    kTok  = kB * kS;                 // 4096 token rows
constexpr size_t kOutElems = (size_t)kTok * kD;   // 4 Mi floats (out tensor)
constexpr float  kInvSqrtDk = 0.125f;             // 1/sqrt(64)
constexpr int    kPB = 20;  // LDS row stride (floats): 8B aligned + conflict-free

__device__ __forceinline__ v8f wmma4(v2f a, v2f b, v8f c) {
  // D = A(16x4,f32) * B(4x16,f32) + C(16x16,f32)
  return __builtin_amdgcn_wmma_f32_16x16x4_f32(false, a, false, b, (short)0, c,
                                               false, false);
}

__device__ __forceinline__ v2f ld2(const float* p) {
  const float2 t = *reinterpret_cast<const float2*>(p);
  v2f r; r.x = t.x; r.y = t.y; return r;
}

// ---------------------------------------------------------------------------
// Kernel 1: per-head QKV projection.  C[4096, h*64+k] = X[4096,1024] @ W'
// W is [H, D, Dk]; a 64-wide N tile == exactly one head's weight slab.
// Block = 4 waves; wave computes 32(M) x 64(N): each B frag feeds 2 WMMAs.
// ---------------------------------------------------------------------------
__global__ __launch_bounds__(128) void qkv_proj_kernel(
    const float* __restrict__ X, const float* __restrict__ W,
    float* __restrict__ out) {
  const int lane = threadIdx.x & 31;
  const int wv   = threadIdx.x >> 5;
  const int lo = lane & 15, hi = lane >> 4;
  const int h  = blockIdx.x;            // 64-col tile == head id
  const int n0 = h * 64;
  const int m0 = blockIdx.y * 128 + wv * 32;

  const float* Arow0 = X + (size_t)(m0 + lo) * kD + 2 * hi;        // M-sub 0
  const float* Arow1 = X + (size_t)(m0 + 16 + lo) * kD + 2 * hi;   // M-sub 1
  const float* Wh    = W + (size_t)h * kD * kDk;                   // [D, Dk]

  const v8f zero = {0.f, 0.f, 0.f, 0.f, 0.f, 0.f, 0.f, 0.f};
  v8f acc[2][4];
#pragma unroll
  for (int r = 0; r < 2; ++r)
#pragma unroll
    for (int t = 0; t < 4; ++t) acc[r][t] = zero;

  for (int kc = 0; kc < kD; kc += 4) {
    const v2f a0 = ld2(Arow0 + kc);                            // A[m][kc+2hi..]
    const v2f a1 = ld2(Arow1 + kc);
    const float* w0 = Wh + (size_t)(kc + 2 * hi) * kDk + lo;   // B[k][n]
#pragma unroll
    for (int t = 0; t < 4; ++t) {
      v2f bf; bf.x = w0[t * 16]; bf.y = w0[kDk + t * 16];
      acc[0][t] = wmma4(a0, bf, acc[0][t]);
      acc[1][t] = wmma4(a1, bf, acc[1][t]);
    }
  }
#pragma unroll
  for (int r = 0; r < 2; ++r)
#pragma unroll
    for (int t = 0; t < 4; ++t)
#pragma unroll
      for (int v = 0; v < 8; ++v)
        out[(size_t)(m0 + r * 16 + v + 8 * hi) * kD + n0 + t * 16 + lo] =
            acc[r][t][v];
}

// ---------------------------------------------------------------------------
// Kernel 2: fused flash-style attention per (h, b, 64-row q-tile).
// Writes masked scaled scores to d_out (required output) and O/l online;
// p_attn never touches HBM.  Wave owns 16 q-rows, Q frags stay in VGPRs.
// ---------------------------------------------------------------------------
__global__ __launch_bounds__(128) void attn_kernel(
    const float* __restrict__ Q, const float* __restrict__ K,
    const float* __restrict__ V, const int* __restrict__ mask,
    float* __restrict__ scores_out, float* __restrict__ concat) {
  __shared__ float pbuf[4][16 * kPB];   // per-wave P-tile transpose buffer

  const int lane = threadIdx.x & 31;
  const int wv   = threadIdx.x >> 5;
  const int lo = lane & 15, hi = lane >> 4;
  const int h = blockIdx.y & (kH - 1);
  const int b = blockIdx.y >> 4;
  const int i0 = blockIdx.x * 64 + wv * 16;     // this wave's q rows

  // Preload all Q A-fragments for this wave's 16 rows (K = 64 -> 16 chunks).
  v2f qa[16];
  {
    const float* qrow = Q + (size_t)(b * kS + i0 + lo) * kD + h * kDk + 2 * hi;
#pragma unroll
    for (int kc = 0; kc < 16; ++kc) qa[kc] = ld2(qrow + kc * 4);
  }

  const v8f zero = {0.f, 0.f, 0.f, 0.f, 0.f, 0.f, 0.f, 0.f};
  v8f accO[4];
#pragma unroll
  for (int t = 0; t < 4; ++t) accO[t] = zero;
  float mrun[8], lrun[8];
#pragma unroll
  for (int v = 0; v < 8; ++v) { mrun[v] = -__builtin_inff(); lrun[v] = 0.f; }

  for (int j0 = 0; j0 < kS; j0 += 16) {
    // ---- S = Q @ K^T (16x16), K^T B-frag is contiguous float2 in K rows ----
    v8f s = zero;
    const float* krow = K + (size_t)(b * kS + j0 + lo) * kD + h * kDk + 2 * hi;
#pragma unroll
    for (int kc = 0; kc < 16; ++kc) s = wmma4(qa[kc], ld2(krow + kc * 4), s);

    // ---- scale, mask, emit masked scores, online softmax update ----
    float p[8];
#pragma unroll
    for (int v = 0; v < 8; ++v) {
      const int row = i0 + v + 8 * hi;          // sequence position i
      const int col = j0 + lo;                  // sequence position j
      float sv = s[v] * kInvSqrtDk;
      if (mask[((size_t)b * kS + row) * kS + col] == 0) sv = -__builtin_inff();
      scores_out[((size_t)(h * kB + b) * kS + row) * kS + col] = sv;

      float rm = sv;                             // row max over 16 lanes
      rm = fmaxf(rm, __shfl_xor(rm, 1, 32));
      rm = fmaxf(rm, __shfl_xor(rm, 2, 32));
      rm = fmaxf(rm, __shfl_xor(rm, 4, 32));
      rm = fmaxf(rm, __shfl_xor(rm, 8, 32));
      const float mnew = fmaxf(mrun[v], rm);
      const float f = __expf(mrun[v] - mnew);
      mrun[v] = mnew;
      p[v] = __expf(sv - mnew);
      float rs = p[v];                           // row sum over 16 lanes
      rs += __shfl_xor(rs, 1, 32);
      rs += __shfl_xor(rs, 2, 32);
      rs += __shfl_xor(rs, 4, 32);
      rs += __shfl_xor(rs, 8, 32);
      lrun[v] = lrun[v] * f + rs;
#pragma unroll
      for (int t = 0; t < 4; ++t) accO[t][v] = accO[t][v] * f;  // rescale O
    }

    // ---- C-layout -> A-layout for P via wave-private LDS tile ----
#pragma unroll
    for (int v = 0; v < 8; ++v) pbuf[wv][(v + 8 * hi) * kPB + lo] = p[v];
    __builtin_amdgcn_wave_barrier();   // LDS is in-order per wave; block reorder

    // ---- O += P(16x16) @ V(16x64) ----
#pragma unroll
    for (int kc = 0; kc < 4; ++kc) {
      const v2f pa = ld2(&pbuf[wv][lo * kPB + kc * 4 + 2 * hi]);
      const float* vrow =
          V + (size_t)(b * kS + j0 + kc * 4 + 2 * hi) * kD + h * kDk + lo;
#pragma unroll
      for (int t = 0; t < 4; ++t) {
        v2f vb; vb.x = vrow[t * 16]; vb.y = vrow[kD + t * 16];
        accO[t] = wmma4(pa, vb, accO[t]);
      }
    }
    __builtin_amdgcn_wave_barrier();   // protect pbuf before next overwrite
  }

  // ---- normalize and store concat[B*S, H*Dk] ----
#pragma unroll
  for (int v = 0; v < 8; ++v) {
    const float inv = 1.0f / lrun[v];
    const int row = i0 + v + 8 * hi;
#pragma unroll
    for (int t = 0; t < 4; ++t)
      concat[(size_t)(b * kS + row) * kD + h * kDk + t * 16 + lo] =
          accO[t][v] * inv;
  }
}

// ---------------------------------------------------------------------------
// Kernel 3: out = concat[4096,1024] @ Wo[1024,1024]  (plain row-major B)
// Wave computes 32x64 like kernel 1.
// ---------------------------------------------------------------------------
__global__ __launch_bounds__(128) void out_proj_kernel(
    const float* __restrict__ X, const float* __restrict__ W,
    float* __restrict__ out) {
  const int lane = threadIdx.x & 31;
  const int wv   = threadIdx.x >> 5;
  const int lo = lane & 15, hi = lane >> 4;
  const int n0 = blockIdx.x * 64;
  const int m0 = blockIdx.y * 128 + wv * 32;

  const float* Arow0 = X + (size_t)(m0 + lo) * kD + 2 * hi;
  const float* Arow1 = X + (size_t)(m0 + 16 + lo) * kD + 2 * hi;

  const v8f zero = {0.f, 0.f, 0.f, 0.f, 0.f, 0.f, 0.f, 0.f};
  v8f acc[2][4];
#pragma unroll
  for (int r = 0; r < 2; ++r)
#pragma unroll
    for (int t = 0; t < 4; ++t) acc[r][t] = zero;

  for (int kc = 0; kc < kD; kc += 4) {
    const v2f a0 = ld2(Arow0 + kc);
    const v2f a1 = ld2(Arow1 + kc);
    const float* w0 = W + (size_t)(kc + 2 * hi) * kD + n0 + lo;
#pragma unroll
    for (int t = 0; t < 4; ++t) {
      v2f bf; bf.x = w0[t * 16]; bf.y = w0[kD + t * 16];
      acc[0][t] = wmma4(a0, bf, acc[0][t]);
      acc[1][t] = wmma4(a1, bf, acc[1][t]);
    }
  }
#pragma unroll
  for (int r = 0; r < 2; ++r)
#pragma unroll
    for (int t = 0; t < 4; ++t)
#pragma unroll
      for (int v = 0; v < 8; ++v)
        out[(size_t)(m0 + r * 16 + v + 8 * hi) * kD + n0 + t * 16 + lo] =
            acc[r][t][v];
}

// ---------------------------------------------------------------------------
extern "C" void kernel_launch(void* const* d_in, const int* in_sizes, int n_in,
                              void* d_out, int out_size, void* d_ws,
                              size_t ws_size, hipStream_t stream) {
  (void)in_sizes; (void)n_in; (void)out_size; (void)ws_size;
  const float* query = (const float*)d_in[0];
  const float* key   = (const float*)d_in[1];
  const float* value = (const float*)d_in[2];
  const int*   mask  = (const int*)d_in[3];
  const float* Wq    = (const float*)d_in[4];
  const float* Wk    = (const float*)d_in[5];
  const float* Wv    = (const float*)d_in[6];
  const float* Wo    = (const float*)d_in[7];

  float* out    = (float*)d_out;            // [B,S,D]
  float* scores = out + kOutElems;          // [H,B,S,S]

  float* qbuf   = (float*)d_ws;             // [B*S, H*Dk] each, 16 MB apiece
  float* kbuf   = qbuf + kOutElems;
  float* vbuf   = kbuf + kOutElems;
  float* concat = vbuf + kOutElems;

  const dim3 blk(128);
  const dim3 gProj(kH, kTok / 128);         // 16 x 32 (N-tiles x M-tiles)
  qkv_proj_kernel<<<gProj, blk, 0, stream>>>(query, Wq, qbuf);
  qkv_proj_kernel<<<gProj, blk, 0, stream>>>(key,   Wk, kbuf);
  qkv_proj_kernel<<<gProj, blk, 0, stream>>>(value, Wv, vbuf);

  const dim3 gAttn(kS / 64, kH * kB);       // 16 x 64 (q-tiles x head*batch)
  attn_kernel<<<gAttn, blk, 0, stream>>>(qbuf, kbuf, vbuf, mask, scores, concat);

  const dim3 gOut(kD / 64, kTok / 128);     // 16 x 32
  out_proj_kernel<<<gOut, blk, 0, stream>>>(concat, Wo, out);
}